// GATv2Layer_77730318123230
// MI455X (gfx1250) — compile-verified
//
#include <hip/hip_runtime.h>
#include <hip/hip_bf16.h>

// GATv2 layer for MI455X (gfx1250), f32 throughout (matches reference dtype).
// GEMM stages: V_WMMA_F32_16X16X4_F32 (wave32 WMMA).
// LDS staging: GLOBAL_LOAD_ASYNC_TO_LDS_B128 + s_wait_asynccnt, double-buffered.
//
//  B=4, N=1024, F_IN=F_OUT=64.

#define B_  4
#define N_  1024
#define F_  64
#define NEG_INF_ -9.0e15f
#define ALPHA_ 0.2f

typedef float v2f __attribute__((ext_vector_type(2)));
typedef float v8f __attribute__((ext_vector_type(8)));

__device__ __forceinline__ v8f wmma_f32(v2f a, v2f b, v8f c) {
  // 8 args: (neg_a, A, neg_b, B, c_mod, C, reuse_a, reuse_b)
  return __builtin_amdgcn_wmma_f32_16x16x4_f32(false, a, false, b, (short)0, c,
                                               false, false);
}

// Stage a 64x64 f32 tile (row-major, row stride 64) into padded LDS [64][68].
// 128 threads, 8 x b128 per thread. On gfx1250: async copy tracked by ASYNCcnt.
__device__ __forceinline__ void stage64x64(float (*dst)[68],
                                           const float* __restrict__ src,
                                           int tid) {
#if defined(__gfx1250__)
#pragma unroll
  for (int t = 0; t < 8; ++t) {
    const int idx = tid + (t << 7);                 // 0..1023 float4 tiles
    const int row = idx >> 4, c4 = (idx & 15) << 2;
    const unsigned int lds = (unsigned int)(size_t)&dst[row][c4];
    const unsigned long long ga =
        (unsigned long long)(size_t)(src + (size_t)row * F_ + c4);
    asm volatile("global_load_async_to_lds_b128 %0, %1, off"
                 :: "v"(lds), "v"(ga) : "memory");
  }
#else
#pragma unroll
  for (int t = 0; t < 8; ++t) {
    const int idx = tid + (t << 7);
    const int row = idx >> 4, c4 = (idx & 15) << 2;
    *(float4*)&dst[row][c4] = *(const float4*)(src + (size_t)row * F_ + c4);
  }
#endif
}

// Wait for staged chunk: async loads complete in order, so asynccnt <= 8 means
// the older stage (8 per-wave ops) is done while the newer 8 remain in flight.
__device__ __forceinline__ void wait_stage(bool more_in_flight) {
#if defined(__gfx1250__)
  if (more_in_flight)
    asm volatile("s_wait_asynccnt 0x8" ::: "memory");
  else
    asm volatile("s_wait_asynccnt 0x0" ::: "memory");
#endif
}

// -----------------------------------------------------------------------------
// Kernel 1: hp = h @ lin_w^T + lin_b ; u = hp @ W1^T ; v = hp @ W2^T
// One wave per 16-row tile. 256 blocks.
//
// WMMA f32 16x16x4 operand layouts (ISA 7.12.2):
//   A (16x4, MxK): lane = M (0..15 both halves); VGPR pair holds K = kh, kh+1
//                  where kh = 2*(lane>=16)  -> contiguous-K v2f load.
//   B (4x16, KxN): lane = N; lane-half picks K = kh, kh+1 (x,y of v2f).
//   C/D (16x16):   VGPR r -> row r + 8*(lane>=16), col = lane&15.
// -----------------------------------------------------------------------------
__global__ __launch_bounds__(32) void gat_k1(const float* __restrict__ h,
                                             const float* __restrict__ lin_w,
                                             const float* __restrict__ lin_b,
                                             const float* __restrict__ Ww,
                                             float* __restrict__ hp,
                                             float* __restrict__ u,
                                             float* __restrict__ v) {
  __shared__ float hps[16][F_];

  const int l  = threadIdx.x;          // 0..31
  const int m  = l & 15;               // M (A) / N (B,C,D) index
  const int kh = (l >> 4) << 1;        // 0 or 2: K sub-offset per lane-half
  const int b  = blockIdx.x >> 6;
  const int i0 = (blockIdx.x & 63) << 4;

  const float* hrow = h + (size_t)(b * N_ + i0 + m) * F_;

  // ---- hp tile: 4 n-tiles of 16x16, chained over K=64 ----
  v8f acc[4] = {v8f{}, v8f{}, v8f{}, v8f{}};
#pragma unroll
  for (int k0 = 0; k0 < F_; k0 += 4) {
    v2f A = *(const v2f*)(hrow + k0 + kh);                 // A[m][k0+kh..+1]
#pragma unroll
    for (int nt = 0; nt < 4; ++nt) {
      // B[k][o] = lin_w[o][k]
      v2f Bv = *(const v2f*)(lin_w + (size_t)(nt * 16 + m) * F_ + k0 + kh);
      acc[nt] = wmma_f32(A, Bv, acc[nt]);
    }
  }

  float* hpg = hp + (size_t)(b * N_ + i0) * F_;
#pragma unroll
  for (int nt = 0; nt < 4; ++nt) {
    const int n = nt * 16 + m;
    const float bias = lin_b[n];
#pragma unroll
    for (int r = 0; r < 8; ++r) {
      const int row = r + ((l >> 4) << 3);
      const float val = acc[nt][r] + bias;
      hps[row][n] = val;
      hpg[(size_t)row * F_ + n] = val;
    }
  }
  __syncthreads();

  // ---- u = hp @ W1^T ; v = hp @ W2^T (A-fragments re-read from LDS) ----
#pragma unroll
  for (int which = 0; which < 2; ++which) {
    const float* W = Ww + which * F_;  // W_w row stride 2*F_
    float* og = (which ? v : u) + (size_t)(b * N_ + i0) * F_;
    v8f d[4] = {v8f{}, v8f{}, v8f{}, v8f{}};
#pragma unroll
    for (int k0 = 0; k0 < F_; k0 += 4) {
      v2f A = *(const v2f*)(&hps[m][k0 + kh]);             // ds_load_b64
#pragma unroll
      for (int nt = 0; nt < 4; ++nt) {
        v2f Bv = *(const v2f*)(W + (size_t)(nt * 16 + m) * (2 * F_) + k0 + kh);
        d[nt] = wmma_f32(A, Bv, d[nt]);
      }
    }
#pragma unroll
    for (int nt = 0; nt < 4; ++nt) {
      const int n = nt * 16 + m;
#pragma unroll
      for (int r = 0; r < 8; ++r) {
        const int row = r + ((l >> 4) << 3);
        og[(size_t)row * F_ + n] = d[nt][r];
      }
    }
  }
}

// -----------------------------------------------------------------------------
// Kernel 2: fused scores + mask + softmax + (attn @ hp) + ELU.
// 128 threads (4 waves) per 16-row tile; e-tile (16x1024) lives in LDS.
// Double-buffered async LDS staging for v-chunks (pass A) and hp-chunks (pass C).
// 256 blocks.
// -----------------------------------------------------------------------------
__global__ __launch_bounds__(128) void gat_k2(const float* __restrict__ hp,
                                              const float* __restrict__ u,
                                              const float* __restrict__ v,
                                              const int* __restrict__ adj,
                                              const float* __restrict__ avec,
                                              float* __restrict__ out) {
  __shared__ float e[16][N_ + 2];     // +2 pad: spreads A-fragment rows over banks
  __shared__ float chunk[2][64][68];  // double-buffered 64-row f32 chunk
  __shared__ float us[16][F_];
  __shared__ float as[F_];
  __shared__ float red[16][8];
  __shared__ float rowmax[16];
  __shared__ float rowinv[16];

  const int tid = threadIdx.x;
  const int b   = blockIdx.x >> 6;
  const int i0  = (blockIdx.x & 63) << 4;

  for (int idx = tid; idx < 16 * F_; idx += 128)
    us[idx >> 6][idx & 63] =
        u[(size_t)(b * N_ + i0 + (idx >> 6)) * F_ + (idx & 63)];
  if (tid < F_) as[tid] = avec[tid];

  const int* adjb   = adj + (size_t)(b * N_ + i0) * N_;
  const float* vb   = v + (size_t)b * N_ * F_;
  const float* hpb  = hp + (size_t)b * N_ * F_;

  // ---------------- Pass A: e[i][j] = sum_f a_f * LeakyReLU(u_if + v_jf) ----
  // Thread owns 8 rows (i = ih..ih+7) x 1 column (jj): the per-lane v-chunk
  // float4 is loaded once and reused 8x; u-row reads are wave-uniform
  // broadcasts.
  const int jj = tid & 63;
  const int ih = (tid >> 6) << 3;     // 0 or 8

  stage64x64(chunk[0], vb, tid);      // prologue: stage jc=0
  for (int jc = 0; jc < 16; ++jc) {
    const int j0 = jc << 6;
    if (jc < 15) stage64x64(chunk[(jc + 1) & 1], vb + (size_t)(j0 + 64) * F_, tid);
    wait_stage(jc < 15);
    __syncthreads();                  // all waves' stage(jc) complete
    float(*cur)[68] = chunk[jc & 1];

    float accq[8] = {0.f, 0.f, 0.f, 0.f, 0.f, 0.f, 0.f, 0.f};
    const float4* cf = (const float4*)&cur[jj][0];
    const float4* af = (const float4*)&as[0];
#pragma unroll
    for (int f4 = 0; f4 < 16; ++f4) {
      const float4 vv = cf[f4];
      const float4 aa = af[f4];
#pragma unroll
      for (int q = 0; q < 8; ++q) {
        const float4 uu = ((const float4*)&us[ih + q][0])[f4];  // broadcast
        float x;
        x = uu.x + vv.x; accq[q] += aa.x * (x > 0.f ? x : ALPHA_ * x);
        x = uu.y + vv.y; accq[q] += aa.y * (x > 0.f ? x : ALPHA_ * x);
        x = uu.z + vv.z; accq[q] += aa.z * (x > 0.f ? x : ALPHA_ * x);
        x = uu.w + vv.w; accq[q] += aa.w * (x > 0.f ? x : ALPHA_ * x);
      }
    }
#pragma unroll
    for (int q = 0; q < 8; ++q) {
      const int i = ih + q;
      e[i][j0 + jj] =
          (adjb[(size_t)i * N_ + j0 + jj] > 0) ? accq[q] : NEG_INF_;
    }
    __syncthreads();                  // buffer may be restaged next iteration
  }

  // ---------------- Pass B: row softmax stats (normalization deferred) -----
  {
    const int rr = tid >> 3, pp = tid & 7;                 // 8 threads per row
    float lmax = -3.0e38f;
    for (int cc = 0; cc < 128; ++cc)
      lmax = fmaxf(lmax, e[rr][pp + (cc << 3)]);
    red[rr][pp] = lmax;
    __syncthreads();
    if (tid < 16) {
      float mx = red[tid][0];
#pragma unroll
      for (int q = 1; q < 8; ++q) mx = fmaxf(mx, red[tid][q]);
      rowmax[tid] = mx;
    }
    __syncthreads();
    const float mx = rowmax[rr];
    float lsum = 0.f;
    for (int cc = 0; cc < 128; ++cc) {
      const int c = pp + (cc << 3);
      const float ex = __expf(e[rr][c] - mx);
      e[rr][c] = ex;                                       // e now holds exp()
      lsum += ex;
    }
    red[rr][pp] = lsum;
    __syncthreads();
    if (tid < 16) {
      float s = 0.f;
#pragma unroll
      for (int q = 0; q < 8; ++q) s += red[tid][q];
      rowinv[tid] = 1.0f / s;                              // fold into epilogue
    }
    __syncthreads();
  }

  // ---------------- Pass C: O(16x64) = P(16x1024) @ hp(1024x64), WMMA chain -
  const int l  = tid & 31;
  const int n0 = (tid >> 5) << 4;                          // wave's 16-col n-tile
  const int m  = l & 15;
  const int kh = (l >> 4) << 1;
  v8f o = {};

  stage64x64(chunk[0], hpb, tid);     // prologue: stage kc=0
  for (int kc = 0; kc < 16; ++kc) {
    const int kbase = kc << 6;
    if (kc < 15)
      stage64x64(chunk[(kc + 1) & 1], hpb + (size_t)(kbase + 64) * F_, tid);
    wait_stage(kc < 15);
    __syncthreads();
    float(*cur)[68] = chunk[kc & 1];
#pragma unroll
    for (int k0 = 0; k0 < 64; k0 += 4) {
      v2f A = *(const v2f*)(&e[m][kbase + k0 + kh]);       // aligned ds_load_b64
      v2f Bv;
      Bv.x = cur[k0 + kh][n0 + m];
      Bv.y = cur[k0 + kh + 1][n0 + m];
      o = wmma_f32(A, Bv, o);
    }
    __syncthreads();
  }

  // ---------------- Epilogue: normalize, ELU, store (D layout) -------------
  float* outb = out + (size_t)(b * N_ + i0) * F_;
#pragma unroll
  for (int r = 0; r < 8; ++r) {
    const int row = r + ((l >> 4) << 3);
    float val = o[r] * rowinv[row];
    val = val > 0.f ? val : expm1f(val);
    outb[(size_t)row * F_ + n0 + m] = val;
  }
}

// -----------------------------------------------------------------------------
extern "C" void kernel_launch(void* const* d_in, const int* in_sizes, int n_in,
                              void* d_out, int out_size, void* d_ws,
                              size_t ws_size, hipStream_t stream) {
  const float* h     = (const float*)d_in[0];
  const int*   adj   = (const int*)d_in[1];
  const float* lin_w = (const float*)d_in[2];
  const float* lin_b = (const float*)d_in[3];
  const float* W_w   = (const float*)d_in[4];
  const float* a     = (const float*)d_in[5];
  float* out = (float*)d_out;

  float* hp = (float*)d_ws;                 // B*N*F floats
  float* u  = hp + (size_t)B_ * N_ * F_;    // B*N*F floats
  float* v  = u + (size_t)B_ * N_ * F_;     // B*N*F floats (3 MB total)

  gat_k1<<<B_ * (N_ / 16), 32, 0, stream>>>(h, lin_w, lin_b, W_w, hp, u, v);
  gat_k2<<<B_ * (N_ / 16), 128, 0, stream>>>(hp, u, v, adj, a, out);
}